// K_CustomLayer_77378130804965
// MI455X (gfx1250) — compile-verified
//
#include <hip/hip_runtime.h>
#include <math.h>

// ---------------------------------------------------------------------------
// DoG-of-Lab, fully fused, one image (128x128x3) per workgroup.
// Separable Gaussian blurs run as banded f32 matmuls on V_WMMA_F32_16X16X4_F32.
// Reflect-padding is folded into precomputed banded weight tables (compact
// [128][2r+1] layout + zero guard column -> branchless fragment builds).
// Wave/tile indices are forced into SGPRs (readfirstlane) so every WMMA loop
// is scalar-controlled and EXEC is all-ones at the matrix ops (ISA req).
// LDS = Lab plane + temp plane + tables (~160 KB of 320 KB/WGP -> 2 WGs).
// ---------------------------------------------------------------------------

typedef __attribute__((ext_vector_type(2))) float v2f;
typedef __attribute__((ext_vector_type(8))) float v8f;

#define STRIDE 129           // LDS row pitch (floats)
#define R1 7
#define L1 15                // sigma 1.65 -> radius int(4*1.65+0.5)=7
#define R2 19
#define L2 39                // sigma 4.85 -> radius int(4*4.85+0.5)=19
#define W1PITCH 16           // 2*R1+1 taps + zero guard col at 15
#define W2PITCH 40           // 2*R2+1 taps + zero guard col at 39

__device__ __forceinline__ float srgb_lin(float v) {
    return v > 0.04045f ? powf((v + 0.055f) * (1.0f / 1.055f), 2.4f)
                        : v * (1.0f / 12.92f);
}
__device__ __forceinline__ float labf(float t) {
    return t > 0.008856f ? cbrtf(t) : 7.787f * t + (16.0f / 116.0f);
}

// One-time build of compact band table W[dst][d] = weight of source column
// j = dst - r + d, with scipy 'reflect' folded in. d in [0, 2r]; columns
// d >= 2r+1 stay zero (guard for out-of-band fragment elements).
__device__ __forceinline__ void build_band(float* __restrict__ W,
                                           const float* __restrict__ k,
                                           int r, int L, int pitch, int tid) {
    for (int idx = tid; idx < 128 * pitch; idx += 256) {
        int dst = idx / pitch;
        int d   = idx - dst * pitch;
        float w = 0.f;
        int j = dst - r + d;
        if (d < L && j >= 0 && j <= 127) {
            w = k[d];                        // direct tap, t == d
            int t = r - 1 - j - dst;         // low reflection (source -1-j)
            if ((unsigned)t < (unsigned)L) w += k[t];
            t = 255 - j - dst + r;           // high reflection (source 255-j)
            if ((unsigned)t < (unsigned)L) w += k[t];
        }
        W[idx] = w;
    }
}

// Vertical blur of one 16x16 tile:  T_tile = W x P   (A = weights, B = data)
// y0/x0 must be wave-uniform (SGPR) so the loop is scalar-controlled.
__device__ __forceinline__ void vblur_tile(const float* __restrict__ P,
                                           float* __restrict__ T,
                                           const float* __restrict__ W,
                                           int r, int pitch,
                                           int y0, int x0, int m, int half) {
    v8f acc = {0.f, 0.f, 0.f, 0.f, 0.f, 0.f, 0.f, 0.f};
    int jlo = (y0 - r) < 0 ? 0 : ((y0 - r) & ~3);
    int jhi = (y0 + 15 + r) > 127 ? 127 : (y0 + 15 + r);
    const int kb   = half << 1;
    const int col  = x0 + m;
    const int dst  = y0 + m;
    const int zcol = 2 * r + 1;              // zero guard column
    const float* Wrow = W + dst * pitch;
    for (int j0 = jlo; j0 <= jhi; j0 += 4) {
        int ja = j0 + kb, jb = ja + 1;
        int da = ja - dst + r, db = da + 1;
        int ia = (ja <= 127 && (unsigned)da <= (unsigned)(2 * r)) ? da : zcol;
        int ib = (jb <= 127 && (unsigned)db <= (unsigned)(2 * r)) ? db : zcol;
        v2f a, b;
        a.x = Wrow[ia];
        a.y = Wrow[ib];
        b.x = P[(ja <= 127 ? ja : 127) * STRIDE + col];
        b.y = P[(jb <= 127 ? jb : 127) * STRIDE + col];
        acc = __builtin_amdgcn_wmma_f32_16x16x4_f32(
                  false, a, false, b, (short)0, acc, false, false);
    }
#pragma unroll
    for (int v = 0; v < 8; ++v)
        T[(y0 + v + 8 * half) * STRIDE + col] = acc[v];
}

// Horizontal blur of one 16x16 tile:  D_tile = T x W^T  (A = data, B = weights)
__device__ __forceinline__ v8f hblur_tile(const float* __restrict__ T,
                                          const float* __restrict__ W,
                                          int r, int pitch,
                                          int y0, int x0, int m, int half) {
    v8f acc = {0.f, 0.f, 0.f, 0.f, 0.f, 0.f, 0.f, 0.f};
    int jlo = (x0 - r) < 0 ? 0 : ((x0 - r) & ~3);
    int jhi = (x0 + 15 + r) > 127 ? 127 : (x0 + 15 + r);
    const int kb   = half << 1;
    const int dst  = x0 + m;                 // output column this lane feeds
    const int zcol = 2 * r + 1;
    const float* rowp = T + (y0 + m) * STRIDE;
    const float* Wrow = W + dst * pitch;
    for (int j0 = jlo; j0 <= jhi; j0 += 4) {
        int ja = j0 + kb, jb = ja + 1;
        int da = ja - dst + r, db = da + 1;
        int ia = (ja <= 127 && (unsigned)da <= (unsigned)(2 * r)) ? da : zcol;
        int ib = (jb <= 127 && (unsigned)db <= (unsigned)(2 * r)) ? db : zcol;
        v2f a, b;
        a.x = rowp[ja <= 127 ? ja : 127];
        a.y = rowp[jb <= 127 ? jb : 127];
        b.x = Wrow[ia];
        b.y = Wrow[ib];
        acc = __builtin_amdgcn_wmma_f32_16x16x4_f32(
                  false, a, false, b, (short)0, acc, false, false);
    }
    return acc;
}

__global__ __launch_bounds__(256)
void dog_lab_wmma_kernel(const float* __restrict__ x, float* __restrict__ out) {
    extern __shared__ float smem[];
    float* P   = smem;                       // 128*STRIDE Lab plane
    float* T   = P + 128 * STRIDE;           // 128*STRIDE vblur temp
    float* W1  = T + 128 * STRIDE;           // 128*16 band table sigma1
    float* W2  = W1 + 128 * W1PITCH;         // 128*40 band table sigma2
    float* K1s = W2 + 128 * W2PITCH;         // 16 taps
    float* K2s = K1s + 16;                   // 40 taps
    float* red = K2s + 40;                   // 512 reduction scratch

    const int tid  = threadIdx.x;
    const int img  = blockIdx.x;
    // Force wave index into an SGPR: makes all tile coords / loop bounds
    // provably uniform -> scalar loops, EXEC all-ones at WMMA.
    const int wave = __builtin_amdgcn_readfirstlane(tid >> 5);
    const int lane = tid & 31;
    const int m    = lane & 15;
    const int half = lane >> 4;
    const int x0   = 16 * wave;              // wave's tile column (SGPR)

    if (tid == 0) {                          // gaussian taps in f64 (scipy)
        double kk[40], s = 0.0;
        for (int t = 0; t < L1; ++t) { double d = (t - R1) / 1.65; kk[t] = exp(-0.5 * d * d); s += kk[t]; }
        for (int t = 0; t < L1; ++t) K1s[t] = (float)(kk[t] / s);
        s = 0.0;
        for (int t = 0; t < L2; ++t) { double d = (t - R2) / 4.85; kk[t] = exp(-0.5 * d * d); s += kk[t]; }
        for (int t = 0; t < L2; ++t) K2s[t] = (float)(kk[t] / s);
    }
    __syncthreads();
    build_band(W1, K1s, R1, L1, W1PITCH, tid);
    build_band(W2, K2s, R2, L2, W2PITCH, tid);
    __syncthreads();

    const float* xin = x   + (size_t)img * (128 * 128 * 3);
    float*       op  = out + (size_t)img * (128 * 128 * 3);

    for (int c = 0; c < 3; ++c) {
        // ---- Lab channel c into P (x re-reads stay resident in 192MB L2) --
        for (int p = tid; p < 128 * 128; p += 256) {
            float r = xin[p * 3 + 0], g = xin[p * 3 + 1], b = xin[p * 3 + 2];
            float lr = srgb_lin(r), lg = srgb_lin(g), lb = srgb_lin(b);
            float fy = labf(0.212671f * lr + 0.715160f * lg + 0.072169f * lb);
            float val;
            if (c == 0) {
                val = 116.0f * fy - 16.0f;
            } else if (c == 1) {
                float fx = labf((0.412453f * lr + 0.357580f * lg + 0.180423f * lb) * (1.0f / 0.95047f));
                val = 500.0f * (fx - fy);
            } else {
                float fz = labf((0.019334f * lr + 0.119193f * lg + 0.950227f * lb) * (1.0f / 1.08883f));
                val = 200.0f * (fy - fz);
            }
            P[(p >> 7) * STRIDE + (p & 127)] = val;
        }
        __syncthreads();

        v8f dog[8];

        // ---- blur1: vertical (-> T), then horizontal (-> regs) ------------
#pragma unroll 1
        for (int i = 0; i < 8; ++i)
            vblur_tile(P, T, W1, R1, W1PITCH, 16 * i, x0, m, half);
        __syncthreads();
#pragma unroll
        for (int i = 0; i < 8; ++i)
            dog[i] = hblur_tile(T, W1, R1, W1PITCH, 16 * i, x0, m, half);
        __syncthreads();

        // ---- blur2: vertical (-> T), horizontal, subtract -----------------
#pragma unroll 1
        for (int i = 0; i < 8; ++i)
            vblur_tile(P, T, W2, R2, W2PITCH, 16 * i, x0, m, half);
        __syncthreads();
#pragma unroll
        for (int i = 0; i < 8; ++i) {
            v8f b2 = hblur_tile(T, W2, R2, W2PITCH, 16 * i, x0, m, half);
#pragma unroll
            for (int v = 0; v < 8; ++v) dog[i][v] -= b2[v];
        }

        // ---- post-process + store ----------------------------------------
        if (c == 0) {
            float mn = 3.402823466e38f, mx = -3.402823466e38f;
#pragma unroll
            for (int i = 0; i < 8; ++i)
#pragma unroll
                for (int v = 0; v < 8; ++v) {
                    mn = fminf(mn, dog[i][v]);
                    mx = fmaxf(mx, dog[i][v]);
                }
            red[tid] = mn; red[256 + tid] = mx;
            __syncthreads();
            for (int s = 128; s > 0; s >>= 1) {
                if (tid < s) {
                    red[tid]       = fminf(red[tid],       red[tid + s]);
                    red[256 + tid] = fmaxf(red[256 + tid], red[256 + tid + s]);
                }
                __syncthreads();
            }
            mn = red[0]; mx = red[256];
            float inv = 1.0f / (mx - mn);
#pragma unroll
            for (int i = 0; i < 8; ++i) {
                int y0 = 16 * i;
#pragma unroll
                for (int v = 0; v < 8; ++v) {
                    int y = y0 + v + 8 * half, xc = x0 + m;
                    op[(y * 128 + xc) * 3 + 0] = (dog[i][v] - mn) * inv;
                }
            }
            __syncthreads();
        } else {
            float scale = (c == 1) ? (1.0f / 10.0f) : 1.0f;
#pragma unroll
            for (int i = 0; i < 8; ++i) {
                int y0 = 16 * i;
#pragma unroll
                for (int v = 0; v < 8; ++v) {
                    int y = y0 + v + 8 * half, xc = x0 + m;
                    op[(y * 128 + xc) * 3 + c] =
                        (dog[i][v] * scale + 128.0f) * (1.0f / 255.0f);
                }
            }
        }
    }
}

extern "C" void kernel_launch(void* const* d_in, const int* in_sizes, int n_in,
                              void* d_out, int out_size, void* d_ws, size_t ws_size,
                              hipStream_t stream) {
    (void)n_in; (void)out_size; (void)d_ws; (void)ws_size;
    const float* x = (const float*)d_in[0];
    float* out = (float*)d_out;
    const int B = in_sizes[0] / (128 * 128 * 3);   // 512 images

    const size_t smem =
        (size_t)(2 * 128 * STRIDE + 128 * W1PITCH + 128 * W2PITCH + 16 + 40 + 512)
        * sizeof(float);
    (void)hipFuncSetAttribute((const void*)dog_lab_wmma_kernel,
                              hipFuncAttributeMaxDynamicSharedMemorySize,
                              (int)smem);
    dog_lab_wmma_kernel<<<B, 256, smem, stream>>>(x, out);
}